// DynamicConv2d_57097295233691
// MI455X (gfx1250) — compile-verified
//
#include <hip/hip_runtime.h>

// ---------------------------------------------------------------------------
// DynamicConv2d for MI455X (gfx1250, wave32, WMMA)
//   B=32, C=64, H=W=128, OUT=64, K=3, HID=16
//   Stage 1: global average pool         -> ws.pooled [32,64]
//   Stage 2: fc1 (relu) + fc2            -> ws.wT     [b][kpos][c][o] (4.7MB)
//   Stage 3: grouped conv as implicit GEMM with V_WMMA_F32_16X16X4_F32.
//            Each wave owns TWO 16-outch tiles sharing one B operand:
//            2 WMMAs per B load, interleaved independent accumulator chains.
// ---------------------------------------------------------------------------

typedef __attribute__((ext_vector_type(2))) float v2f;
typedef __attribute__((ext_vector_type(8))) float v8f;

#define BB   32
#define CC   64
#define HH   128
#define WW2  128
#define OUTC 64
#define HID  16
#define K9   9
#define HW   (HH * WW2)          // 16384
#define ATILE (K9 * CC * 16)     // 9216 floats per 16-outch weight tile

// ---------------- Stage 1: global average pool -----------------------------
__global__ __launch_bounds__(256)
void pool_kernel(const float* __restrict__ x, float* __restrict__ pooled) {
    __shared__ float red[256];
    const int bc = blockIdx.x;                     // 0 .. B*C-1
    const float* p = x + (size_t)bc * HW;
    float s = 0.f;
    for (int i = threadIdx.x; i < HW; i += 256) s += p[i];
    red[threadIdx.x] = s;
    __syncthreads();
    for (int off = 128; off > 0; off >>= 1) {
        if ((int)threadIdx.x < off) red[threadIdx.x] += red[threadIdx.x + off];
        __syncthreads();
    }
    if (threadIdx.x == 0) pooled[bc] = red[0] * (1.0f / HW);
}

// ---------------- Stage 2a: h = relu(pooled @ w1^T + b1) --------------------
__global__ __launch_bounds__(512)
void fc1_kernel(const float* __restrict__ pooled, const float* __restrict__ w1,
                const float* __restrict__ b1, float* __restrict__ h) {
    const int t = threadIdx.x;                     // 512 = B*HID threads
    if (t >= BB * HID) return;
    const int b = t / HID, j = t % HID;
    const float* pr = pooled + b * CC;
    const float* wr = w1 + j * CC;
    float s = b1[j];
    #pragma unroll 8
    for (int c = 0; c < CC; ++c) s += pr[c] * wr[c];
    h[t] = s > 0.f ? s : 0.f;
}

// ---------------- Stage 2b: wgen = h @ w2^T + b2, stored transposed ---------
// wT layout: [b][kpos][c][o]   (K-major-friendly for the conv GEMM)
__global__ __launch_bounds__(256)
void fc2_kernel(const float* __restrict__ h, const float* __restrict__ w2,
                const float* __restrict__ b2, float* __restrict__ wT) {
    long t = (long)blockIdx.x * 256 + threadIdx.x;
    const long TOT = (long)BB * OUTC * CC * K9;    // 1,179,648
    if (t >= TOT) return;
    const int kpos = (int)(t % K9);  long r = t / K9;
    const int c    = (int)(r % CC);  r /= CC;
    const int o    = (int)(r % OUTC);
    const int b    = (int)(r / OUTC);
    const int rr   = (o * CC + c) * K9 + kpos;     // row of w2 / wgen
    const float* hb = h + b * HID;
    const float* wr = w2 + (long)rr * HID;
    float s = b2[rr];
    #pragma unroll
    for (int j = 0; j < HID; ++j) s += hb[j] * wr[j];
    wT[(((size_t)b * K9 + kpos) * CC + c) * OUTC + o] = s;
}

// ---------------- Stage 3: grouped conv as WMMA implicit GEMM ---------------
// Block = (b, mth in 0..1, y in 0..127); 8 waves; wave w handles pixels
// x0 = 16*w .. 16*w+15 of row y, output channels [mth*32, mth*32+32).
//
// VGPR layouts (ISA 7.12.2, wave32):
//   A 16x4 f32 : lane -> M = lane%16, vgpr v -> K = 2*(lane/16) + v
//   B 4x16 f32 : lane -> N = lane%16, vgpr v -> K = 2*(lane/16) + v
//   D 16x16 f32: vgpr r -> M = r + 8*(lane/16), N = lane%16
//
// LDS A layout packs channel pairs contiguously per M so each A fetch is one
// ds_load_b64:  ldsA[tile*ATILE + ((kpos*32 + k/2)*16 + m)*2 + (k&1)]
__global__ __launch_bounds__(256)
void conv_wmma_kernel(const float* __restrict__ x,
                      const float* __restrict__ wT,
                      float* __restrict__ out) {
    __shared__ float ldsA[2 * ATILE];              // 18432 floats = 72 KB

    int blk = blockIdx.x;
    const int y   = blk % HH;  blk /= HH;
    const int mth = blk % 2;   blk /= 2;
    const int b   = blk;                           // 0..31
    const int mt0 = mth * 2;                       // first 16-outch tile index

    // Cooperative stage of the two (b, mt) weight tiles into LDS.
    for (int idx = threadIdx.x; idx < 2 * ATILE; idx += 256) {
        const int m    = idx & 15;
        int rest       = idx >> 4;                 // t*576 + kpos*64 + c
        const int c    = rest & 63;
        rest >>= 6;                                // t*9 + kpos
        const int kpos = rest % K9;
        const int t    = rest / K9;                // 0/1
        const float v  = wT[(((size_t)b * K9 + kpos) * CC + c) * OUTC
                            + (mt0 + t) * 16 + m];
        ldsA[t * ATILE + ((kpos * 32 + (c >> 1)) * 16 + m) * 2 + (c & 1)] = v;
    }
    __syncthreads();

    const int lane = threadIdx.x & 31;
    const int wav  = threadIdx.x >> 5;             // 0..7
    const int hl   = lane >> 4;                    // 0/1
    const int lq   = lane & 15;                    // 0..15
    const int x0   = wav * 16;

    v8f acc0 = {};
    v8f acc1 = {};
    const float* xb = x + (size_t)b * CC * HW;

    for (int kpos = 0; kpos < K9; ++kpos) {
        const int ky = kpos / 3 - 1;
        const int kx = kpos % 3 - 1;
        const int yy = y + ky;
        if (yy < 0 || yy >= HH) continue;          // uniform across the wave
        const int  xx  = x0 + lq + kx;
        const bool inb = (xx >= 0) && (xx < WW2);  // per-lane -> select, no div.
        const float* xrow = xb + (size_t)yy * WW2 + (inb ? xx : 0);
        const float* aRow0 = &ldsA[kpos * 1024];           // tile 0, this kpos
        const float* aRow1 = &ldsA[ATILE + kpos * 1024];   // tile 1, this kpos

        #pragma unroll
        for (int c = 0; c < CC; c += 4) {
            const int k0 = c + 2 * hl;             // even
            // One B operand, shared by both output-channel tiles.
            v2f bv;
            const float b0 = xrow[(size_t)k0 * HW];
            const float b1 = xrow[(size_t)(k0 + 1) * HW];
            bv.x = inb ? b0 : 0.0f;
            bv.y = inb ? b1 : 0.0f;
            const int aoff = (((k0 >> 1) * 16) + lq) * 2;
            const v2f av0 = *(const v2f*)&aRow0[aoff];
            const v2f av1 = *(const v2f*)&aRow1[aoff];
            // Two independent accumulator chains: consecutive WMMAs never
            // share a D register, covering the WMMA->WMMA hazard window.
            acc0 = __builtin_amdgcn_wmma_f32_16x16x4_f32(
                       false, av0, false, bv, (short)0, acc0, false, false);
            acc1 = __builtin_amdgcn_wmma_f32_16x16x4_f32(
                       false, av1, false, bv, (short)0, acc1, false, false);
        }
    }

    // Scatter D: vgpr r -> outch = (mt0+t)*16 + 8*hl + r, pixel = (y, x0+lq)
    const size_t pixBase = (size_t)y * WW2 + x0 + lq;
    const int ob0 = mt0 * 16 + 8 * hl;
    const size_t base0 = ((size_t)b * OUTC + ob0) * HW + pixBase;
    const size_t base1 = base0 + (size_t)16 * HW;  // next 16-outch tile
    #pragma unroll
    for (int r = 0; r < 8; ++r) {
        out[base0 + (size_t)r * HW] = acc0[r];
        out[base1 + (size_t)r * HW] = acc1[r];
    }
}

// ---------------------------------------------------------------------------
extern "C" void kernel_launch(void* const* d_in, const int* in_sizes, int n_in,
                              void* d_out, int out_size, void* d_ws, size_t ws_size,
                              hipStream_t stream) {
    const float* x  = (const float*)d_in[0];
    const float* w1 = (const float*)d_in[1];
    const float* b1 = (const float*)d_in[2];
    const float* w2 = (const float*)d_in[3];
    const float* b2 = (const float*)d_in[4];
    float* out = (float*)d_out;

    // Workspace: pooled[2048] | h[512] | wT[1,179,648]  (~4.73 MB total)
    float* pooled = (float*)d_ws;
    float* h      = pooled + BB * CC;
    float* wT     = h + BB * HID;

    pool_kernel<<<BB * CC, 256, 0, stream>>>(x, pooled);
    fc1_kernel<<<1, 512, 0, stream>>>(pooled, w1, b1, h);

    const long tot = (long)BB * OUTC * CC * K9;    // 1,179,648
    fc2_kernel<<<(int)((tot + 255) / 256), 256, 0, stream>>>(h, w2, b2, wT);

    conv_wmma_kernel<<<BB * 2 * HH, 256, 0, stream>>>(x, wT, out);
}